// SparseLieBracket_61409442398926
// MI455X (gfx1250) — compile-verified
//
#include <hip/hip_runtime.h>
#include <stdint.h>

// z[b,k] = sum_e x[b,I[e]] * y[b,J[e]] * C[e]
// B=2048, DIM=248, E=60000. LDS-resident gather/scatter, lane==batch.

#define LIE_DIM   248
#define LANES     32
#define SEGMENTS  16
#define LDS_FLOATS (3 * LIE_DIM * LANES)
#define LDS_BYTES  (LDS_FLOATS * 4)

// ---- gfx1250 async global->LDS copy (ASYNCcnt path, CDNA5 ISA) ----
// GV mode: VDST holds the LDS byte offset from the workgroup's LDS base; VADDR holds the 64-bit global address.
// Per-lane LDS destination gives a free row-major -> [d][lane] transpose while
// the async engine runs ahead of the wave.
__device__ __forceinline__ void async_f32_to_lds(uint32_t lds_byte_off,
                                                 const float* gptr) {
  asm volatile("global_load_async_to_lds_b32 %0, %1, off"
               :: "v"(lds_byte_off), "v"(gptr)
               : "memory");
}

__device__ __forceinline__ void wait_async_zero() {
#if defined(__has_builtin) && __has_builtin(__builtin_amdgcn_s_wait_asynccnt)
  __builtin_amdgcn_s_wait_asynccnt(0);
#else
  asm volatile("s_wait_asynccnt 0" ::: "memory");
#endif
}

__device__ __forceinline__ void global_fadd(float* p, float v) {
#if defined(__has_builtin) && __has_builtin(__builtin_amdgcn_global_atomic_fadd_f32)
  (void)__builtin_amdgcn_global_atomic_fadd_f32(p, v);
#else
  unsafeAtomicAdd(p, v);
#endif
}

__global__ __launch_bounds__(256)
void lie_zero_kernel(float* __restrict__ out, int n) {
  int i = blockIdx.x * blockDim.x + threadIdx.x;
  if (i < n) out[i] = 0.0f;
}

__global__ __launch_bounds__(LANES)
void lie_bracket_kernel(const float* __restrict__ x,
                        const float* __restrict__ y,
                        const int*   __restrict__ I,
                        const int*   __restrict__ J,
                        const int*   __restrict__ K,
                        const float* __restrict__ C,
                        float* __restrict__ z,
                        int nEntries) {
  extern __shared__ float lds[];
  float* xs = lds;                       // [LIE_DIM][LANES]
  float* ys = lds + LIE_DIM * LANES;     // [LIE_DIM][LANES]
  float* zs = lds + 2 * LIE_DIM * LANES; // [LIE_DIM][LANES]

  const int lane  = threadIdx.x;         // 0..31  (wave32: lane == sub-batch)
  const int bg    = blockIdx.x;          // batch group (32 batches each)
  const int seg   = blockIdx.y;          // entry segment
  const int batch = bg * LANES + lane;

  // Zero this workgroup's z accumulators (single wave -> no barrier needed;
  // DS ops are in-order within the wave).
  for (int i = lane; i < LIE_DIM * LANES; i += LANES) zs[i] = 0.0f;

  // Async-transpose x,y rows for our 32 batches into [d][lane] LDS columns.
  // Uniform d across the wave -> 32 consecutive LDS banks -> conflict-free.
  const float* xrow = x + (size_t)batch * LIE_DIM;
  const float* yrow = y + (size_t)batch * LIE_DIM;
  for (int d = 0; d < LIE_DIM; ++d) {
    uint32_t off = (uint32_t)((d * LANES + lane) * 4);
    async_f32_to_lds(off,                       xrow + d);
    async_f32_to_lds(off + LIE_DIM * LANES * 4, yrow + d);
  }

  // Segment bounds: rounded to a multiple of 4 so the SMEM index stream can be
  // read as aligned 16B (s_load_b128) packets.
  const int per = (((nEntries + SEGMENTS - 1) / SEGMENTS) + 3) & ~3;
  const int e0  = (seg * per < nEntries) ? seg * per : nEntries;
  const int e1  = (e0 + per < nEntries) ? (e0 + per) : nEntries;

  // While ASYNCcnt drains, warm the index stream for this segment into cache
  // with gfx1250 global_prefetch_b8 (lanes striped one 64B line apart).
  {
    const int segBytes = (e1 - e0) * 4;
    const char* pI = (const char*)(I + e0);
    const char* pJ = (const char*)(J + e0);
    const char* pK = (const char*)(K + e0);
    const char* pC = (const char*)(C + e0);
    for (int off = lane * 64; off < segBytes; off += LANES * 64) {
      __builtin_prefetch(pI + off, 0, 0);
      __builtin_prefetch(pJ + off, 0, 0);
      __builtin_prefetch(pK + off, 0, 0);
      __builtin_prefetch(pC + off, 0, 0);
    }
  }

  wait_async_zero();

  // Hot loop: 4 entries per iteration. Uniform 16B index reads -> s_load_b128
  // broadcast across the wave; per entry: 2 conflict-free ds_load_b32 gathers,
  // 2 VALU mults, 1 non-returning ds_add_f32 scatter (fire-and-forget; LDS
  // keeps same-address updates ordered within the wave).
  const int    nQuad = (e1 - e0) >> 2;
  const int4*  I4 = (const int4*)(I + e0);
  const int4*  J4 = (const int4*)(J + e0);
  const int4*  K4 = (const int4*)(K + e0);
  const float4* C4 = (const float4*)(C + e0);

  for (int t = 0; t < nQuad; ++t) {
    const int4   iv = I4[t];
    const int4   jv = J4[t];
    const int4   kv = K4[t];
    const float4 cv = C4[t];
    atomicAdd(&zs[kv.x * LANES + lane],
              xs[iv.x * LANES + lane] * ys[jv.x * LANES + lane] * cv.x);
    atomicAdd(&zs[kv.y * LANES + lane],
              xs[iv.y * LANES + lane] * ys[jv.y * LANES + lane] * cv.y);
    atomicAdd(&zs[kv.z * LANES + lane],
              xs[iv.z * LANES + lane] * ys[jv.z * LANES + lane] * cv.z);
    atomicAdd(&zs[kv.w * LANES + lane],
              xs[iv.w * LANES + lane] * ys[jv.w * LANES + lane] * cv.w);
  }
  for (int e = e0 + (nQuad << 2); e < e1; ++e) {
    const int i = I[e], j = J[e], k = K[e];
    atomicAdd(&zs[k * LANES + lane],
              xs[i * LANES + lane] * ys[j * LANES + lane] * C[e]);
  }

  // Combine segment partials into global z with hardware f32 atomics.
  float* zrow = z + (size_t)batch * LIE_DIM;
  for (int d = 0; d < LIE_DIM; ++d)
    global_fadd(zrow + d, zs[d * LANES + lane]);
}

extern "C" void kernel_launch(void* const* d_in, const int* in_sizes, int n_in,
                              void* d_out, int out_size, void* d_ws, size_t ws_size,
                              hipStream_t stream) {
  const float* x = (const float*)d_in[0];
  const float* y = (const float*)d_in[1];
  const int*   I = (const int*)d_in[2];
  const int*   J = (const int*)d_in[3];
  const int*   K = (const int*)d_in[4];
  const float* C = (const float*)d_in[5];
  float* z = (float*)d_out;

  const int nEntries = in_sizes[2];
  const int B        = in_sizes[0] / LIE_DIM;   // 2048

  // Zero output (segment partials are atomically accumulated into it).
  lie_zero_kernel<<<(out_size + 255) / 256, 256, 0, stream>>>(z, out_size);

  // 93 KB dynamic LDS per (single-wave) workgroup; raise the per-kernel cap.
  hipFuncSetAttribute((const void*)lie_bracket_kernel,
                      hipFuncAttributeMaxDynamicSharedMemorySize, LDS_BYTES);

  dim3 grid(B / LANES, SEGMENTS, 1);   // 64 batch-groups x 16 entry segments
  dim3 block(LANES, 1, 1);             // one wave32 per workgroup
  lie_bracket_kernel<<<grid, block, LDS_BYTES, stream>>>(x, y, I, J, K, C, z,
                                                         nEntries);
}